// DeepFM_56075093016733
// MI455X (gfx1250) — compile-verified
//
#include <hip/hip_runtime.h>
#include <hip/hip_bf16.h>

typedef __attribute__((ext_vector_type(2))) float v2f;
typedef __attribute__((ext_vector_type(8))) float v8f;

#define BATCH      16384
#define NUM_FIELDS 26
#define VOCAB      100000
#define EMBED_DIM  32
#define DEEP_IN    832      // NUM_FIELDS * EMBED_DIM
#define LN_EPS     1e-5f

// ---------------------------------------------------------------------------
// Kernel 1: embedding gather + FM first/second order.
// One wave32 per batch row; lane index == embedding dim (D == 32 == warpSize).
// Writes embeds (output #2, also the MLP input x) and fm_out[row] to ws.
// ---------------------------------------------------------------------------
__global__ __launch_bounds__(256) void fm_gather_kernel(
    const int* __restrict__ idx,        // [B, F]
    const float* __restrict__ emb,      // [F, V, D]
    const float* __restrict__ bias,     // [F, V, 1]
    float* __restrict__ embeds,         // [B, F*D]  (== x)
    float* __restrict__ fm)             // [B]
{
  const int lane = threadIdx.x & 31;
  const int row  = (int)((blockIdx.x * blockDim.x + threadIdx.x) >> 5);
  if (row >= BATCH) return;

  const int* irow = idx + (size_t)row * NUM_FIELDS;
  float* erow = embeds + (size_t)row * DEEP_IN;

  float s = 0.f, ssq = 0.f;
#pragma unroll
  for (int f = 0; f < NUM_FIELDS; ++f) {
    const int ix = irow[f];                                  // uniform (SGPR) load
    const float v = emb[((size_t)f * VOCAB + ix) * EMBED_DIM + lane]; // coalesced 128B
    erow[f * EMBED_DIM + lane] = v;
    s   += v;
    ssq += v * v;
  }

  // first-order: lane f (<26) fetches its bias; reduce with second-order term.
  float bsum = (lane < NUM_FIELDS) ? bias[(size_t)lane * VOCAB + irow[lane]] : 0.f;
  float red  = s * s - ssq;  // per-dim contribution to 2nd order
#pragma unroll
  for (int off = 16; off > 0; off >>= 1) {
    red  += __shfl_down(red,  off, 32);
    bsum += __shfl_down(bsum, off, 32);
  }
  if (lane == 0) fm[row] = bsum + 0.5f * red;
}

// ---------------------------------------------------------------------------
// Kernel 2: Y = LayerNorm(ReLU(X @ W + b)) * g + beta
// One wave32 per 16-row stripe; the wave owns all N columns (N = 128/64/32),
// so the LayerNorm reduction over N never leaves the wave.
// Matrix core: V_WMMA_F32_16X16X4_F32, K-loop in steps of 4.
//   A frag (16x4 f32): lane L holds row M=L%16, cols k+2*(L/16)+{0,1}  (2 VGPRs)
//   B frag (4x16 f32): lane L holds col N=L%16, rows k+2*(L/16)+{0,1}  (2 VGPRs)
//   C/D   (16x16 f32): VGPR v, lane L -> M = v + 8*(L/16), N = L%16    (8 VGPRs)
// ---------------------------------------------------------------------------
template <int N>
__global__ __launch_bounds__(32) void gemm_relu_ln_kernel(
    const float* __restrict__ X,     // [B, K]
    const float* __restrict__ W,     // [K, N]
    const float* __restrict__ b,     // [N]
    const float* __restrict__ g,     // [N]
    const float* __restrict__ beta,  // [N]
    float* __restrict__ Y,           // [B, N]
    int K)
{
  constexpr int NT = N / 16;
  __shared__ float tile[16 * N];

  const int lane  = (int)threadIdx.x;   // 0..31, EXEC all-ones through WMMA
  const int lmod  = lane & 15;
  const int lhalf = lane >> 4;
  const int m0    = (int)blockIdx.x * 16;

  v8f acc[NT] = {};

  const float* arow = X + (size_t)(m0 + lmod) * K + 2 * lhalf;
  for (int k = 0; k < K; k += 4) {
    v2f a;
    a.x = arow[k];
    a.y = arow[k + 1];
    const float* bptr = W + (size_t)(k + 2 * lhalf) * N + lmod;
#pragma unroll
    for (int t = 0; t < NT; ++t) {
      v2f bf;
      bf.x = bptr[t * 16];       // row k+2*lhalf
      bf.y = bptr[t * 16 + N];   // row k+2*lhalf+1
      acc[t] = __builtin_amdgcn_wmma_f32_16x16x4_f32(
          /*neg_a=*/false, a, /*neg_b=*/false, bf,
          /*c_mod=*/(short)0, acc[t], /*reuse_a=*/false, /*reuse_b=*/false);
    }
  }

  // bias + ReLU -> LDS tile [16][N]
#pragma unroll
  for (int t = 0; t < NT; ++t) {
#pragma unroll
    for (int v = 0; v < 8; ++v) {
      const int m = v + 8 * lhalf;
      const int n = t * 16 + lmod;
      tile[m * N + n] = fmaxf(acc[t][v] + b[n], 0.f);
    }
  }
  __syncthreads();  // single-wave WG: ordering fence for LDS

  // LayerNorm: lanes 0..15 each normalize one full row.
  if (lane < 16) {
    float mu = 0.f, sq = 0.f;
#pragma unroll 4
    for (int n = 0; n < N; ++n) {
      const float h = tile[lane * N + n];
      mu += h;
      sq += h * h;
    }
    mu *= (1.f / N);
    const float var = sq * (1.f / N) - mu * mu;
    const float inv = rsqrtf(var + LN_EPS);
    float* yrow = Y + (size_t)(m0 + lane) * N;
#pragma unroll 4
    for (int n = 0; n < N; ++n)
      yrow[n] = g[n] * (tile[lane * N + n] - mu) * inv + beta[n];
  }
}

// ---------------------------------------------------------------------------
// Kernel 3: out = sigmoid(fm + h2 @ Wout + bout).  Wave per row (H2 == 32).
// ---------------------------------------------------------------------------
__global__ __launch_bounds__(256) void final_kernel(
    const float* __restrict__ h,     // [B, 32]
    const float* __restrict__ Wout,  // [32, 1]
    const float* __restrict__ bout,  // [1]
    const float* __restrict__ fm,    // [B]
    float* __restrict__ out)         // [B]
{
  const int lane = threadIdx.x & 31;
  const int row  = (int)((blockIdx.x * blockDim.x + threadIdx.x) >> 5);
  if (row >= BATCH) return;

  float v = h[(size_t)row * 32 + lane] * Wout[lane];
#pragma unroll
  for (int off = 16; off > 0; off >>= 1) v += __shfl_down(v, off, 32);

  if (lane == 0) {
    const float z = fm[row] + v + bout[0];
    out[row] = 1.f / (1.f + expf(-z));
  }
}

// ---------------------------------------------------------------------------
extern "C" void kernel_launch(void* const* d_in, const int* in_sizes, int n_in,
                              void* d_out, int out_size, void* d_ws, size_t ws_size,
                              hipStream_t stream) {
  const int*   idx   = (const int*)  d_in[0];
  const float* emb   = (const float*)d_in[1];
  const float* bias  = (const float*)d_in[2];
  const float* W0    = (const float*)d_in[3];
  const float* b0    = (const float*)d_in[4];
  const float* g0    = (const float*)d_in[5];
  const float* beta0 = (const float*)d_in[6];
  const float* W1    = (const float*)d_in[7];
  const float* b1    = (const float*)d_in[8];
  const float* g1    = (const float*)d_in[9];
  const float* beta1 = (const float*)d_in[10];
  const float* W2    = (const float*)d_in[11];
  const float* b2    = (const float*)d_in[12];
  const float* g2    = (const float*)d_in[13];
  const float* beta2 = (const float*)d_in[14];
  const float* Wout  = (const float*)d_in[15];
  const float* bout  = (const float*)d_in[16];

  // d_out = [ output(B) | embeds(B*F*D) ]; embeds doubles as MLP input x.
  float* out    = (float*)d_out;
  float* embeds = out + BATCH;

  // workspace: fm | h0 | h1 | h2
  float* fm = (float*)d_ws;
  float* h0 = fm + BATCH;
  float* h1 = h0 + (size_t)BATCH * 128;
  float* h2 = h1 + (size_t)BATCH * 64;

  fm_gather_kernel<<<BATCH / 8, 256, 0, stream>>>(idx, emb, bias, embeds, fm);

  gemm_relu_ln_kernel<128><<<BATCH / 16, 32, 0, stream>>>(embeds, W0, b0, g0, beta0, h0, DEEP_IN);
  gemm_relu_ln_kernel<64> <<<BATCH / 16, 32, 0, stream>>>(h0,     W1, b1, g1, beta1, h1, 128);
  gemm_relu_ln_kernel<32> <<<BATCH / 16, 32, 0, stream>>>(h1,     W2, b2, g2, beta2, h2, 64);

  final_kernel<<<BATCH / 8, 256, 0, stream>>>(h2, Wout, bout, fm, out);
}